// ModelNew_33921651704535
// MI455X (gfx1250) — compile-verified
//
#include <hip/hip_runtime.h>

typedef float v2f __attribute__((ext_vector_type(2)));
typedef float v4f __attribute__((ext_vector_type(4)));
typedef float v8f __attribute__((ext_vector_type(8)));

#define N_IMG 32
#define H_DIM 128
#define W_DIM 128
#define C_IN  64
#define C_OUT 64

// LDS layout (floats):
//   xs : [4][130][64]       input rows h0-1..h0+2 with zero-padded W halo
//   wl : [9][32][64][2]     wl[kk][c2][n][j] = w[n][kk][2*c2+j]   (kk = kh*3+kw)
#define XS_FLOATS (4 * 130 * 64)
#define WL_FLOATS (9 * 32 * 64 * 2)
#define SMEM_BYTES ((XS_FLOATS + WL_FLOATS) * sizeof(float))

extern "C" __global__ void __launch_bounds__(128, 1)
conv3x3_bias_res_relu_wmma(const float* __restrict__ x,
                           const float* __restrict__ w,
                           const float* __restrict__ bias,
                           const float* __restrict__ identity,
                           float* __restrict__ out)
{
    extern __shared__ float smem[];
    float* xs = smem;               // [4][130][64]
    float* wl = smem + XS_FLOATS;   // [9][32][64][2]

    const int tid = threadIdx.x;
    const int bx  = blockIdx.x;
    const int n   = bx >> 6;          // image index
    const int h0  = (bx & 63) << 1;   // first of two output rows

    // ---- Phase 1a: 4 input rows (h0-1 .. h0+2) into LDS ----
    // Interior: async global->LDS b128 (ASYNCcnt-tracked, no VGPR staging).
    // Halo (padding rows/cols): plain ds_store_b128 zeros (disjoint slots).
    for (int i = tid; i < 4 * 130 * 16; i += 128) {
        const int c4   = i & 15;
        const int t    = i >> 4;
        const int wcol = t % 130;       // 0..129  (input w = wcol-1)
        const int r    = t / 130;       // 0..3    (input h = h0 + r - 1)
        const int h_in = h0 + r - 1;
        const int w_in = wcol - 1;
        const int xidx = (r * 130 + wcol) * 64 + c4 * 4;
        if ((unsigned)h_in < (unsigned)H_DIM && (unsigned)w_in < (unsigned)W_DIM) {
            const float* gp = x + ((((size_t)n * H_DIM + h_in) * W_DIM + w_in) * C_IN + c4 * 4);
            const unsigned lds_off = (unsigned)(xidx * 4);  // byte offset in LDS alloc
            asm volatile("global_load_async_to_lds_b128 %0, %1, off"
                         :: "v"(lds_off), "v"(gp) : "memory");
        } else {
            *(v4f*)(xs + xidx) = (v4f){0.f, 0.f, 0.f, 0.f};
        }
    }

    // ---- Phase 1b: load + transpose weights into LDS ----
    // w layout: [cout][kh][kw][cin] = [64][9][64]; wl[kk][c2][n][0..1]
    for (int i = tid; i < 9 * 32 * 64; i += 128) {
        const int nch = i & 63;
        const int t   = i >> 6;
        const int c2  = t & 31;
        const int kk  = t >> 5;
        const float* gp = w + (((size_t)nch * 9 + kk) * 64 + 2 * c2);
        v2f val = *(const v2f*)gp;
        *(v2f*)(wl + (size_t)i * 2) = val;
    }

    // All async copies from this wave must land before the barrier.
    asm volatile("s_wait_asynccnt 0x0" ::: "memory");
    __syncthreads();

    // ---- Phase 2: implicit-GEMM K loop, 64 pixels x 64 channels per wave ----
    // 4 waves: waveId = {row, half}; each wave owns a 64-pixel strip of one row.
    const int lane    = tid & 31;
    const int waveId  = tid >> 5;
    const int mlane   = lane & 15;    // M row (A) / N col (B)
    const int klane   = lane >> 4;    // 0: K=0,1   1: K=2,3
    const int row_sel = waveId >> 1;  // which of the two output rows
    const int wbase   = (waveId & 1) * 64;  // 64-pixel strip
    const int wpix0   = wbase + mlane;

    v8f acc[4][4];
#pragma unroll
    for (int mt = 0; mt < 4; ++mt)
#pragma unroll
        for (int nt = 0; nt < 4; ++nt)
            acc[mt][nt] = (v8f){0.f,0.f,0.f,0.f,0.f,0.f,0.f,0.f};

#pragma unroll
    for (int kk = 0; kk < 9; ++kk) {
        const int kh = kk / 3;
        const int kw = kk % 3;
        // A base: xs[row_sel + kh][wpix + kw][*]  (halo folded: col = w_in+1)
        const float* arow = xs + (((row_sel + kh) * 130) + (wpix0 + kw)) * 64;
        const float* wrow = wl + (size_t)(kk * 32) * 64 * 2;

#pragma unroll
        for (int c = 0; c < 64; c += 4) {
            // A fragments: four 16x4 f32 tiles (pixel groups wbase + 16*mt)
            v2f a0 = *(const v2f*)(arow + 0 * 1024 + c + klane * 2);
            v2f a1 = *(const v2f*)(arow + 1 * 1024 + c + klane * 2);
            v2f a2 = *(const v2f*)(arow + 2 * 1024 + c + klane * 2);
            v2f a3 = *(const v2f*)(arow + 3 * 1024 + c + klane * 2);

            const int c2 = (c >> 1) + klane;
            const float* bbase = wrow + (size_t)c2 * 128 + mlane * 2;
            // B fragments: 4x16 f32 per 16-channel tile
            v2f b0 = *(const v2f*)(bbase + 0);
            v2f b1 = *(const v2f*)(bbase + 32);
            v2f b2 = *(const v2f*)(bbase + 64);
            v2f b3 = *(const v2f*)(bbase + 96);

            acc[0][0] = __builtin_amdgcn_wmma_f32_16x16x4_f32(false, a0, false, b0, (short)0, acc[0][0], false, false);
            acc[0][1] = __builtin_amdgcn_wmma_f32_16x16x4_f32(false, a0, false, b1, (short)0, acc[0][1], false, false);
            acc[0][2] = __builtin_amdgcn_wmma_f32_16x16x4_f32(false, a0, false, b2, (short)0, acc[0][2], false, false);
            acc[0][3] = __builtin_amdgcn_wmma_f32_16x16x4_f32(false, a0, false, b3, (short)0, acc[0][3], false, false);
            acc[1][0] = __builtin_amdgcn_wmma_f32_16x16x4_f32(false, a1, false, b0, (short)0, acc[1][0], false, false);
            acc[1][1] = __builtin_amdgcn_wmma_f32_16x16x4_f32(false, a1, false, b1, (short)0, acc[1][1], false, false);
            acc[1][2] = __builtin_amdgcn_wmma_f32_16x16x4_f32(false, a1, false, b2, (short)0, acc[1][2], false, false);
            acc[1][3] = __builtin_amdgcn_wmma_f32_16x16x4_f32(false, a1, false, b3, (short)0, acc[1][3], false, false);
            acc[2][0] = __builtin_amdgcn_wmma_f32_16x16x4_f32(false, a2, false, b0, (short)0, acc[2][0], false, false);
            acc[2][1] = __builtin_amdgcn_wmma_f32_16x16x4_f32(false, a2, false, b1, (short)0, acc[2][1], false, false);
            acc[2][2] = __builtin_amdgcn_wmma_f32_16x16x4_f32(false, a2, false, b2, (short)0, acc[2][2], false, false);
            acc[2][3] = __builtin_amdgcn_wmma_f32_16x16x4_f32(false, a2, false, b3, (short)0, acc[2][3], false, false);
            acc[3][0] = __builtin_amdgcn_wmma_f32_16x16x4_f32(false, a3, false, b0, (short)0, acc[3][0], false, false);
            acc[3][1] = __builtin_amdgcn_wmma_f32_16x16x4_f32(false, a3, false, b1, (short)0, acc[3][1], false, false);
            acc[3][2] = __builtin_amdgcn_wmma_f32_16x16x4_f32(false, a3, false, b2, (short)0, acc[3][2], false, false);
            acc[3][3] = __builtin_amdgcn_wmma_f32_16x16x4_f32(false, a3, false, b3, (short)0, acc[3][3], false, false);
        }
    }

    // ---- Phase 3: fused bias + residual + ReLU epilogue ----
    // D layout: VGPR r -> (M=r, N=lane) lanes 0-15 ; (M=r+8, N=lane-16) lanes 16-31
    const int h = h0 + row_sel;
    const size_t pixbase = ((size_t)n * H_DIM + h) * W_DIM;
#pragma unroll
    for (int mt = 0; mt < 4; ++mt) {
#pragma unroll
        for (int nt = 0; nt < 4; ++nt) {
            const int nch = mlane + nt * 16;
            const float bv = bias[nch];
#pragma unroll
            for (int r = 0; r < 8; ++r) {
                const int m = r + klane * 8;
                const size_t gi = (pixbase + (size_t)(wbase + mt * 16 + m)) * C_OUT + nch;
                float v = acc[mt][nt][r] + bv + identity[gi];
                out[gi] = v > 0.f ? v : 0.f;
            }
        }
    }
}

extern "C" void kernel_launch(void* const* d_in, const int* in_sizes, int n_in,
                              void* d_out, int out_size, void* d_ws, size_t ws_size,
                              hipStream_t stream) {
    (void)in_sizes; (void)n_in; (void)out_size; (void)d_ws; (void)ws_size;
    const float* x        = (const float*)d_in[0];
    const float* w        = (const float*)d_in[1];
    const float* bias     = (const float*)d_in[2];
    const float* identity = (const float*)d_in[3];
    float* out            = (float*)d_out;

    hipFuncSetAttribute((const void*)conv3x3_bias_res_relu_wmma,
                        hipFuncAttributeMaxDynamicSharedMemorySize,
                        (int)SMEM_BYTES);

    conv3x3_bias_res_relu_wmma<<<N_IMG * (H_DIM / 2), 128, SMEM_BYTES, stream>>>(
        x, w, bias, identity, out);
}